// CircuitModule_50027779064069
// MI455X (gfx1250) — compile-verified
//
#include <hip/hip_runtime.h>

#define N_VARS  4096
#define N_LAYERS 12
#define WIDTH   8192
#define BATCH   1024

typedef __attribute__((ext_vector_type(2))) float v2f;
typedef __attribute__((ext_vector_type(8))) float v8f;

// Ping-pong node-major value buffers: vals[node][batch], 2 x 32 MB.
__device__ __align__(16) float g_buf[2][(size_t)WIDTH * BATCH];

// ---------------------------------------------------------------------------
// Init: transpose weights (1024 x 4096, batch-major) into node-major layout,
// nodes [0,4096) = w, nodes [4096,8192) = 1 - w  (neg_weights recomputed).
// LDS-tiled 32x32 transpose, fully coalesced both directions.
// ---------------------------------------------------------------------------
__global__ __launch_bounds__(256) void init_transpose(const float* __restrict__ w) {
    __shared__ float tile[32][33];
    const int tx = threadIdx.x;           // 0..31
    const int ty = threadIdx.y;           // 0..7
    const int v0 = blockIdx.x * 32;       // var tile
    const int b0 = blockIdx.y * 32;       // batch tile

#pragma unroll
    for (int i = 0; i < 4; ++i) {
        const int b = b0 + ty + i * 8;
        tile[ty + i * 8][tx] = w[(size_t)b * N_VARS + v0 + tx];
    }
    __syncthreads();
#pragma unroll
    for (int i = 0; i < 4; ++i) {
        const int v = v0 + ty + i * 8;
        const float x = tile[tx][ty + i * 8];
        g_buf[0][(size_t)v * BATCH + b0 + tx]            = x;
        g_buf[0][(size_t)(v + N_VARS) * BATCH + b0 + tx] = 1.0f - x;
    }
}

// ---------------------------------------------------------------------------
// One layer: block = one output node, 256 threads x float4 = 1024 batch lanes.
// Child indices are wave-uniform -> scalar loads; row reads are contiguous.
// ---------------------------------------------------------------------------
template <int PROD>
__global__ __launch_bounds__(256) void layer_kernel(const int* __restrict__ ch, int src) {
    const int w  = blockIdx.x;
    const int c0 = ch[2 * w];
    const int c1 = ch[2 * w + 1];
    const float4* __restrict__ s = (const float4*)g_buf[src];
    float4* __restrict__ d = (float4*)g_buf[src ^ 1];
    const int t = threadIdx.x;

    const float4 x = s[(size_t)c0 * (BATCH / 4) + t];
    const float4 y = s[(size_t)c1 * (BATCH / 4) + t];
    float4 r;
    if (PROD) {
        r.x = x.x * y.x; r.y = x.y * y.y; r.z = x.z * y.z; r.w = x.w * y.w;
    } else {
        r.x = x.x + y.x; r.y = x.y + y.y; r.z = x.z + y.z; r.w = x.w + y.w;
    }
    d[(size_t)w * (BATCH / 4) + t] = r;
}

// ---------------------------------------------------------------------------
// Final reduction as WMMA GEMV: out(1x1024) = ones(1x8192) * vals(8192x1024).
// V_WMMA_F32_16X16X4_F32, A = ones(16x4), f32 accumulate.
// 8 blocks x 8 waves; each wave owns 16 batch columns, K-loop over 8192 rows
// in steps of 16 using 4 independent accumulators (breaks D->C RAW chain).
// With A == 1, the result is exact for any K-row permutation in the B layout;
// columns map as N = lane % 16 (documented B/C/D striping).
// ---------------------------------------------------------------------------
__global__ __launch_bounds__(256) void reduce_wmma(float* __restrict__ out, int src) {
    const float* __restrict__ vals = g_buf[src];
    const int lane   = threadIdx.x & 31;
    const int wave   = threadIdx.x >> 5;
    const int b0     = (blockIdx.x * 8 + wave) * 16;   // 16 batch columns / wave
    const int col    = b0 + (lane & 15);
    const int rowoff = lane >> 4;
    const float* __restrict__ p = vals + (size_t)rowoff * BATCH + col;

    const v2f a = {1.0f, 1.0f};          // A matrix: all ones
    v8f acc0 = {}, acc1 = {}, acc2 = {}, acc3 = {};

    for (int k = 0; k < WIDTH; k += 16) {
        v2f bv;
        bv.x = p[(size_t)(k + 0)  * BATCH];
        bv.y = p[(size_t)(k + 2)  * BATCH];
        acc0 = __builtin_amdgcn_wmma_f32_16x16x4_f32(false, a, false, bv,
                                                     (short)0, acc0, false, false);
        bv.x = p[(size_t)(k + 4)  * BATCH];
        bv.y = p[(size_t)(k + 6)  * BATCH];
        acc1 = __builtin_amdgcn_wmma_f32_16x16x4_f32(false, a, false, bv,
                                                     (short)0, acc1, false, false);
        bv.x = p[(size_t)(k + 8)  * BATCH];
        bv.y = p[(size_t)(k + 10) * BATCH];
        acc2 = __builtin_amdgcn_wmma_f32_16x16x4_f32(false, a, false, bv,
                                                     (short)0, acc2, false, false);
        bv.x = p[(size_t)(k + 12) * BATCH];
        bv.y = p[(size_t)(k + 14) * BATCH];
        acc3 = __builtin_amdgcn_wmma_f32_16x16x4_f32(false, a, false, bv,
                                                     (short)0, acc3, false, false);
    }

    v8f t0 = acc0 + acc1;
    v8f t1 = acc2 + acc3;
    v8f ct = t0 + t1;
    // Every D row holds the column sums; take row M=0 (VGPR 0, lanes 0..15).
    if (lane < 16) out[b0 + lane] = ct[0];
}

// ---------------------------------------------------------------------------
extern "C" void kernel_launch(void* const* d_in, const int* in_sizes, int n_in,
                              void* d_out, int out_size, void* d_ws, size_t ws_size,
                              hipStream_t stream) {
    (void)in_sizes; (void)n_in; (void)out_size; (void)d_ws; (void)ws_size;
    const float* weights  = (const float*)d_in[0];   // (1024, 4096)
    // d_in[1] (neg_weights) is recomputed on the fly as 1 - w.
    const int*   children = (const int*)d_in[2];     // (12, 8192, 2)
    float*       out      = (float*)d_out;           // (1024,)

    init_transpose<<<dim3(N_VARS / 32, BATCH / 32), dim3(32, 8), 0, stream>>>(weights);

    for (int l = 0; l < N_LAYERS; ++l) {
        const int* ch = children + (size_t)l * WIDTH * 2;
        if ((l & 1) == 0)
            layer_kernel<1><<<WIDTH, 256, 0, stream>>>(ch, l & 1);
        else
            layer_kernel<0><<<WIDTH, 256, 0, stream>>>(ch, l & 1);
    }
    // After layer 11 (src=1 -> dst=0), final values live in g_buf[0].
    reduce_wmma<<<8, 256, 0, stream>>>(out, 0);
}